// RNN_lstm_52810917872233
// MI455X (gfx1250) — compile-verified
//
#include <hip/hip_runtime.h>
#include <hip/hip_bf16.h>

typedef __attribute__((ext_vector_type(2))) float v2f;
typedef __attribute__((ext_vector_type(8))) float v8f;

#define T_DIM 512
#define B_DIM 64
#define I_DIM 256
#define H_DIM 512
#define O_DIM 256

// d_out layout (floats): [outputs | Hs | Cs]
#define OUT_HS_OFF (T_DIM * B_DIM * O_DIM)                  // 8388608
#define OUT_CS_OFF (OUT_HS_OFF + T_DIM * B_DIM * H_DIM)     // 25165824

#define NBLOCKS_REC 64
#define NTHREADS    256   // 8 wave32 waves per workgroup

// LDS A-tile strides: (K + 4) floats -> stride mod 64 == 4 -> all 32 lanes'
// ds_load_b64 fragments hit 64 distinct banks (conflict-free).
#define SH_STRIDE (H_DIM + 4)    // 516
#define SX_STRIDE (I_DIM + 4)    // 260

// workspace layout (float offsets; barrier lives in first 256 bytes)
#define WS_S_OFF   64                                   // S: [4][B][H]
#define WS_PX_OFF  (WS_S_OFF  + 4 * B_DIM * H_DIM)      // packed W_x: 4*I*H
#define WS_PH_OFF  (WS_PX_OFF + 4 * I_DIM * H_DIM)      // packed W_h: 4*H*H
#define WS_PQ_OFF  (WS_PH_OFF + 4 * H_DIM * H_DIM)      // packed W_hq: H*O
#define WS_END_OFF (WS_PQ_OFF + H_DIM * O_DIM)          // ~7.1 MB total

// ---------------------------------------------------------------------------
// fp32 WMMA: D = A(16x4) * B(4x16) + C on the matrix pipe.
// args: (neg_a, A, neg_b, B, c_mod, C, reuse_a, reuse_b)
// ---------------------------------------------------------------------------
__device__ __forceinline__ v8f wmma_f32_16x16x4(v2f a, v2f b, v8f c) {
    return __builtin_amdgcn_wmma_f32_16x16x4_f32(false, a, false, b,
                                                 (short)0, c, false, false);
}

// ---------------------------------------------------------------------------
// Device-wide sense-reversing barrier (all NBLOCKS_REC blocks resident)
// ---------------------------------------------------------------------------
__device__ __forceinline__ void grid_sync(unsigned* bar) {
    __threadfence();            // release
    __syncthreads();
    if (threadIdx.x == 0) {
        volatile unsigned* gen = bar + 1;
        unsigned g = *gen;
        if (atomicAdd(bar, 1u) == NBLOCKS_REC - 1) {
            atomicExch(bar, 0u);
            atomicAdd((unsigned*)(bar + 1), 1u);
        } else {
            while (*gen == g) { __builtin_amdgcn_s_sleep(2); }
        }
    }
    __syncthreads();
    __threadfence();            // acquire
}

__global__ void init_barrier_kernel(unsigned* bar) {
    if (threadIdx.x < 2) bar[threadIdx.x] = 0u;
}

// ---------------------------------------------------------------------------
// Repack a [K,N] row-major weight into B-fragment order:
//   P[c][q][lane][j] = W[4q + 2*(lane/16) + j][c*16 + lane%16],  j in {0,1}
// ---------------------------------------------------------------------------
__global__ void repack_kernel(const float* __restrict__ W, float* __restrict__ P,
                              int K, int N) {
    const int idx = blockIdx.x * blockDim.x + threadIdx.x;
    const int total = (K * N) >> 1;
    if (idx >= total) return;
    const int l    = idx & 31;
    const int rest = idx >> 5;
    const int q    = rest % (K >> 2);
    const int c    = rest / (K >> 2);
    const int k    = 4 * q + 2 * (l >> 4);
    const int n    = c * 16 + (l & 15);
    P[2 * (size_t)idx + 0] = W[(size_t)(k + 0) * N + n];
    P[2 * (size_t)idx + 1] = W[(size_t)(k + 1) * N + n];
}

// ---------------------------------------------------------------------------
// Persistent LSTM recurrence. 512 waves; all 8 waves of a block share the
// same (gate g, batch block r) and differ in hidden block c, so the block
// cooperatively stages its 16 A rows (X[t] and H[t-1]) into LDS once per
// step; WMMA A-fragments then come from ds_load_b64 (conflict-free layout)
// while B-fragments stream from the packed weights with global_load_b64.
// ---------------------------------------------------------------------------
__global__ void __launch_bounds__(NTHREADS) lstm_rec_kernel(
    const float* __restrict__ X,
    const float* __restrict__ Px,   // packed W_x, 4 gates
    const float* __restrict__ Ph,   // packed W_h, 4 gates
    const float* __restrict__ bi,  const float* __restrict__ bf_,
    const float* __restrict__ bo,  const float* __restrict__ bg,
    float* __restrict__ out, unsigned* __restrict__ bar,
    float* __restrict__ S)
{
    __shared__ float sH[16 * SH_STRIDE];   // 33,024 B
    __shared__ float sX[16 * SX_STRIDE];   // 16,640 B

    float* Hs = out + OUT_HS_OFF;
    float* Cs = out + OUT_CS_OFF;

    const int tid  = threadIdx.x;
    const int lane = tid & 31;
    // block-uniform tile coordinates
    const int g  = blockIdx.x >> 4;               // gate 0..3
    const int r  = (blockIdx.x >> 2) & 3;         // batch block 0..3
    const int c  = ((blockIdx.x & 3) << 3) + (tid >> 5);   // hidden block 0..31

    const int m0   = r * 16;                 // batch row base of tile
    const int koff = (lane >> 4) * 2;        // A-fragment K sub-offset
    const int ncol = c * 16 + (lane & 15);   // C column for this lane
    const int mst  = (lane >> 4) << 3;       // C-fragment M offset (0 or 8)

    // LDS A-fragment bases for this lane
    const float* aX = &sX[(lane & 15) * SX_STRIDE + koff];
    const float* aH = &sH[(lane & 15) * SH_STRIDE + koff];

    // packed-B stream bases (sequential, coalesced b64 reads)
    const float* PBx = Px + (size_t)g * I_DIM * H_DIM
                          + ((size_t)c * (I_DIM / 4)) * 64 + lane * 2;
    const float* PBh = Ph + (size_t)g * H_DIM * H_DIM
                          + ((size_t)c * (H_DIM / 4)) * 64 + lane * 2;

    const int gtid = blockIdx.x * NTHREADS + tid;        // 0..16383
    float* Sg = S + (size_t)g * B_DIM * H_DIM;

    for (int t = 0; t < T_DIM; ++t) {
        // ------------- cooperative A-tile staging (coalesced) -------------
        {
            const float4* Xg = (const float4*)(X + (size_t)t * B_DIM * I_DIM
                                                 + (size_t)m0 * I_DIM);
            #pragma unroll
            for (int i = tid; i < 16 * (I_DIM / 4); i += NTHREADS) {
                const int row = i >> 6, c4 = i & 63;             // I/4 == 64
                *(float4*)&sX[row * SX_STRIDE + c4 * 4] =
                    Xg[row * (I_DIM / 4) + c4];
            }
            if (t > 0) {
                const float4* Hg = (const float4*)(Hs
                    + (size_t)(t - 1) * B_DIM * H_DIM + (size_t)m0 * H_DIM);
                #pragma unroll
                for (int i = tid; i < 16 * (H_DIM / 4); i += NTHREADS) {
                    const int row = i >> 7, c4 = i & 127;        // H/4 == 128
                    *(float4*)&sH[row * SH_STRIDE + c4 * 4] =
                        Hg[row * (H_DIM / 4) + c4];
                }
            }
        }
        __syncthreads();

        // ---------------- GEMM phase (fp32 WMMA) ----------------
        v8f acc = {};   // biases added in pointwise phase

        #pragma unroll 8
        for (int q = 0; q < I_DIM / 4; ++q) {           // x-side, K = 256
            v2f a = *(const v2f*)(aX + 4 * q);
            v2f b = *(const v2f*)(PBx + (size_t)q * 64);
            acc = wmma_f32_16x16x4(a, b, acc);
        }
        if (t > 0) {
            #pragma unroll 8
            for (int q = 0; q < H_DIM / 4; ++q) {       // h-side, K = 512
                v2f a = *(const v2f*)(aH + 4 * q);
                v2f b = *(const v2f*)(PBh + (size_t)q * 64);
                acc = wmma_f32_16x16x4(a, b, acc);
            }
        }
        // scatter C fragment to staging: S[g][m][n]
        #pragma unroll
        for (int v = 0; v < 8; ++v) {
            Sg[(size_t)(m0 + mst + v) * H_DIM + ncol] = acc[v];
        }

        // warm caches with next step's X tile (16 rows * 1KB = 256 lines)
        if (t + 1 < T_DIM) {
            const char* nx = (const char*)(X + (size_t)(t + 1) * B_DIM * I_DIM
                                             + (size_t)m0 * I_DIM);
            __builtin_prefetch(nx + tid * 64, 0, 1);
        }

        grid_sync(bar);

        // ---------------- pointwise LSTM cell ----------------
        #pragma unroll
        for (int rep = 0; rep < 2; ++rep) {
            const int e = gtid + rep * (NBLOCKS_REC * NTHREADS);  // 0..B*H-1
            const int h = e & (H_DIM - 1);
            const float xi = S[0 * B_DIM * H_DIM + e] + bi[h];
            const float xf = S[1 * B_DIM * H_DIM + e] + bf_[h];
            const float xo = S[2 * B_DIM * H_DIM + e] + bo[h];
            const float xg = S[3 * B_DIM * H_DIM + e] + bg[h];
            const float cp = (t == 0) ? 0.0f
                             : Cs[(size_t)(t - 1) * B_DIM * H_DIM + e];
            const float ig = 1.0f / (1.0f + __expf(-xi));
            const float fg = 1.0f / (1.0f + __expf(-xf));
            const float og = 1.0f / (1.0f + __expf(-xo));
            const float gg = tanhf(xg);
            const float cn = fg * cp + ig * gg;
            const float hn = og * tanhf(cn);
            Cs[(size_t)t * B_DIM * H_DIM + e] = cn;
            Hs[(size_t)t * B_DIM * H_DIM + e] = hn;
        }

        grid_sync(bar);   // Hs[t]/Cs[t] + LDS reuse safe before next step
    }
}

// ---------------------------------------------------------------------------
// outputs = Hs @ W_hq + b_q : [32768,512] x [512,256], one 16x16 tile / wave
// ---------------------------------------------------------------------------
__global__ void __launch_bounds__(NTHREADS) out_proj_kernel(
    const float* __restrict__ Hs, const float* __restrict__ Pq,
    const float* __restrict__ bq, float* __restrict__ Y)
{
    const int tid  = threadIdx.x;
    const int lane = tid & 31;
    const int wid  = blockIdx.x * (NTHREADS / 32) + (tid >> 5);  // 0..32767
    const int r    = wid >> 4;                                   // 0..2047
    const int c    = wid & 15;                                   // 0..15

    const int m0   = r * 16;
    const int mrow = m0 + (lane & 15);
    const int koff = (lane >> 4) * 2;
    const int ncol = c * 16 + (lane & 15);
    const int mst  = (lane >> 4) << 3;

    const float bias = bq[ncol];
    v8f acc;
    #pragma unroll
    for (int v = 0; v < 8; ++v) acc[v] = bias;   // C init = bias(n)

    const float* Arow = Hs + (size_t)mrow * H_DIM;
    const float* PB   = Pq + ((size_t)c * (H_DIM / 4)) * 64 + lane * 2;
    #pragma unroll 8
    for (int q = 0; q < H_DIM / 4; ++q) {
        v2f a = *(const v2f*)(Arow + 4 * q + koff);
        v2f b = *(const v2f*)(PB + (size_t)q * 64);
        acc = wmma_f32_16x16x4(a, b, acc);
    }
    #pragma unroll
    for (int v = 0; v < 8; ++v) {
        Y[(size_t)(m0 + mst + v) * O_DIM + ncol] = acc[v];
    }
}

// ---------------------------------------------------------------------------
extern "C" void kernel_launch(void* const* d_in, const int* in_sizes, int n_in,
                              void* d_out, int out_size, void* d_ws, size_t ws_size,
                              hipStream_t stream) {
    const float* X   = (const float*)d_in[0];
    const float* Wx[4] = { (const float*)d_in[1], (const float*)d_in[2],
                           (const float*)d_in[3], (const float*)d_in[4] };
    const float* Wh[4] = { (const float*)d_in[5], (const float*)d_in[6],
                           (const float*)d_in[7], (const float*)d_in[8] };
    const float* bi  = (const float*)d_in[9];
    const float* bf_ = (const float*)d_in[10];
    const float* bo  = (const float*)d_in[11];
    const float* bg  = (const float*)d_in[12];
    const float* Whq = (const float*)d_in[13];
    const float* bq  = (const float*)d_in[14];

    float*    wsf = (float*)d_ws;
    unsigned* bar = (unsigned*)d_ws;
    float*    S   = wsf + WS_S_OFF;
    float*    Px  = wsf + WS_PX_OFF;
    float*    Ph  = wsf + WS_PH_OFF;
    float*    Pq  = wsf + WS_PQ_OFF;

    // zero grid-barrier state every call (ws poisoned to 0xAA once)
    init_barrier_kernel<<<1, 32, 0, stream>>>(bar);

    // repack all weights into B-fragment streaming order (once per call)
    for (int g = 0; g < 4; ++g) {
        repack_kernel<<<(I_DIM * H_DIM / 2) / 256, 256, 0, stream>>>(
            Wx[g], Px + (size_t)g * I_DIM * H_DIM, I_DIM, H_DIM);
        repack_kernel<<<(H_DIM * H_DIM / 2) / 256, 256, 0, stream>>>(
            Wh[g], Ph + (size_t)g * H_DIM * H_DIM, H_DIM, H_DIM);
    }
    repack_kernel<<<(H_DIM * O_DIM / 2) / 256, 256, 0, stream>>>(
        Whq, Pq, H_DIM, O_DIM);

    // persistent recurrence: 64 WGs x 8 waves = 512 waves, 1 tile/wave/step
    lstm_rec_kernel<<<NBLOCKS_REC, NTHREADS, 0, stream>>>(
        X, Px, Ph, bi, bf_, bo, bg, (float*)d_out, bar, S);

    // output projection: 32768 tiles, 8 waves/block -> 4096 blocks
    const int nblk = (T_DIM * B_DIM / 16) * (O_DIM / 16) / (NTHREADS / 32);
    out_proj_kernel<<<nblk, NTHREADS, 0, stream>>>(
        (float*)d_out + OUT_HS_OFF, Pq, bq, (float*)d_out);
}